// AFNOLayer_34548716929255
// MI455X (gfx1250) — compile-verified
//
#include <hip/hip_runtime.h>
#include <hip/hip_bf16.h>

// ---------------------------------------------------------------------------
// AFNO layer for MI455X (gfx1250, wave32).
//   x[4,128,128,768] -> rfft2(ortho) -> blockwise complex MLP (WMMA f32)
//   -> irfft2(ortho) -> + skip
// Frequency data kept as separate re/im f32 planes [B*H*Fw, D] in d_ws.
// Tile staging uses CDNA5 async global->LDS copies (ASYNCcnt).
// ---------------------------------------------------------------------------

#define Bz 4
#define Hh 128
#define Ww 128
#define Dd 768
#define Fw 65
#define NBLK 8
#define BS 96               // block size == hidden size
#define TOTAL_M (Bz * Hh * Fw)   // 33280
#define CG 32               // channels per FFT workgroup tile
#define NDG (Dd / CG)       // 24 channel groups
#define LSTR 100            // padded LDS row stride for MLP tiles (conflict-free)
#define THRESH 0.01f
#define INV128 0.0078125f

typedef float v2f __attribute__((ext_vector_type(2)));
typedef float v8f __attribute__((ext_vector_type(8)));

// D = A(16x4) * B(4x16) + C(16x16), all f32, wave32.
__device__ __forceinline__ v8f wmma4(v2f a, v2f b, v8f c) {
    return __builtin_amdgcn_wmma_f32_16x16x4_f32(
        /*neg_a=*/false, a, /*neg_b=*/false, b,
        /*c_mod=*/(short)0, c, /*reuse_a=*/false, /*reuse_b=*/false);
}

// Low 32 bits of a generic pointer to a __shared__ object == LDS byte offset
// (ISA: LDS aperture addresses map to LDS via addr[31:0]).
__device__ __forceinline__ unsigned lds_off(const void* p) {
    return (unsigned)(unsigned long long)p;
}

// Async DMA: 16 bytes global -> LDS, tracked by ASYNCcnt.
__device__ __forceinline__ void async_ld_b128(unsigned lds_byte_off, const void* g) {
    asm volatile("global_load_async_to_lds_b128 %0, %1, off"
                 :: "v"(lds_byte_off), "v"(g) : "memory");
}
// Async DMA: 16 bytes LDS -> global.
__device__ __forceinline__ void async_st_b128(void* g, unsigned lds_byte_off) {
    asm volatile("global_store_async_from_lds_b128 %0, %1, off"
                 :: "v"(g), "v"(lds_byte_off) : "memory");
}
__device__ __forceinline__ void async_wait0() {
    asm volatile("s_wait_asynccnt 0x0" ::: "memory");
}

// ---------------------------------------------------------------------------
// In-place radix-2 DIT FFT core: 128 points x CG channels resident in LDS.
// Data must be loaded in bit-reversed order.  256 threads: thread t handles
// channel c = t&31 and butterfly slot j = t>>5 (8 butterflies per stage).
// ---------------------------------------------------------------------------
__device__ __forceinline__ void fft_core(float* sr, float* si, int t, float sgn) {
    const int c = t & (CG - 1);
    const int j = t >> 5;               // 0..7
    for (int s = 0; s < 7; ++s) {
        __syncthreads();
        const int half = 1 << s;
        const float astep = sgn * 3.14159265358979f / (float)half;
#pragma unroll
        for (int q = 0; q < 8; ++q) {
            const int bIdx = j * 8 + q;             // 0..63
            const int g = bIdx >> s;
            const int k = bIdx & (half - 1);
            const int i0 = (g << (s + 1)) + k;
            const int i1 = i0 + half;
            float cw, sw;
            __sincosf(astep * (float)k, &sw, &cw);
            const int a0 = i0 * CG + c;
            const int a1 = i1 * CG + c;
            const float xr = sr[a1], xi = si[a1];
            const float tr = cw * xr - sw * xi;
            const float ti = cw * xi + sw * xr;
            const float ur = sr[a0], ui = si[a0];
            sr[a1] = ur - tr;  si[a1] = ui - ti;
            sr[a0] = ur + tr;  si[a0] = ui + ti;
        }
    }
    __syncthreads();
}

// ---------------------------------------------------------------------------
// Stage A: forward rFFT along W.  Folds the full 2D ortho scale (1/128) at
// the (VALU) store so the nonlinear MLP sees ortho-normalized coefficients.
// grid = (B*H) * NDG, 256 threads, 32KB LDS.  Input tile staged with async
// global->LDS B128 copies (bit-reversed row scatter, contiguous channels).
// ---------------------------------------------------------------------------
__global__ __launch_bounds__(256)
void fft_w_fwd(const float* __restrict__ x, float* __restrict__ Fr,
               float* __restrict__ Fi) {
    __shared__ __align__(16) float sr[Ww * CG];
    __shared__ __align__(16) float si[Ww * CG];
    const int wg = blockIdx.x;
    const int dg = wg % NDG;
    const int bh = wg / NDG;                 // b*H + h
    const int t  = threadIdx.x;
    const float* src = x + (size_t)bh * Ww * Dd + dg * CG;
    // async-stage the real input, bit-reversed along w
    for (int idx = t; idx < Ww * (CG / 4); idx += 256) {
        const int w  = idx >> 3;
        const int c4 = (idx & 7) << 2;
        const int wb = __brev((unsigned)w) >> 25;      // 7-bit bit reversal
        async_ld_b128(lds_off(&sr[wb * CG + c4]), src + (size_t)w * Dd + c4);
    }
    // zero the imaginary plane with plain LDS stores
    for (int idx = t; idx < Ww * CG; idx += 256) si[idx] = 0.0f;
    async_wait0();
    fft_core(sr, si, t, -1.0f);
    float* dr = Fr + (size_t)bh * Fw * Dd + dg * CG;
    float* di = Fi + (size_t)bh * Fw * Dd + dg * CG;
    for (int idx = t; idx < Fw * CG; idx += 256) {
        const int f = idx >> 5, c = idx & (CG - 1);
        dr[(size_t)f * Dd + c] = sr[f * CG + c] * INV128;
        di[(size_t)f * Dd + c] = si[f * CG + c] * INV128;
    }
}

// ---------------------------------------------------------------------------
// Stages B/D: complex FFT along H, in place (sgn = -1 fwd, +1 inv).
// grid = (B*Fw) * NDG, 256 threads.  Loads AND stores use the async
// global<->LDS B128 path (pure data movement, no VALU work needed).
// ---------------------------------------------------------------------------
__global__ __launch_bounds__(256)
void fft_h(float* __restrict__ Fr, float* __restrict__ Fi, float sgn) {
    __shared__ __align__(16) float sr[Hh * CG];
    __shared__ __align__(16) float si[Hh * CG];
    const int wg = blockIdx.x;
    const int dg = wg % NDG;
    const int bf = wg / NDG;
    const int b = bf / Fw, f = bf % Fw;
    const int t = threadIdx.x;
    const size_t base = ((size_t)b * Hh * Fw + f) * Dd + dg * CG;
    const size_t hstr = (size_t)Fw * Dd;
    for (int idx = t; idx < Hh * (CG / 4); idx += 256) {
        const int h  = idx >> 3;
        const int c4 = (idx & 7) << 2;
        const int hb = __brev((unsigned)h) >> 25;
        async_ld_b128(lds_off(&sr[hb * CG + c4]), Fr + base + (size_t)h * hstr + c4);
        async_ld_b128(lds_off(&si[hb * CG + c4]), Fi + base + (size_t)h * hstr + c4);
    }
    async_wait0();
    fft_core(sr, si, t, sgn);
    for (int idx = t; idx < Hh * (CG / 4); idx += 256) {
        const int h  = idx >> 3;
        const int c4 = (idx & 7) << 2;
        async_st_b128(Fr + base + (size_t)h * hstr + c4, lds_off(&sr[h * CG + c4]));
        async_st_b128(Fi + base + (size_t)h * hstr + c4, lds_off(&si[h * CG + c4]));
    }
    async_wait0();   // S_ENDPGM would wait anyway; make the intent explicit
}

// ---------------------------------------------------------------------------
// Stage C: blockwise complex MLP in the frequency domain, in place.
//   h = CReLU(Z W1^T + b1);  y = softshrink(h W2^T + b2)
// Workgroup = 6 waves (192 thr): one 16-row M-tile of one block; wave w owns
// the 16-wide N-tile [16w,16w+16).  K accumulated with f32 WMMA 16x16x4.
// Z tile staged with async global->LDS B128 copies.
// ---------------------------------------------------------------------------
__global__ __launch_bounds__(192)
void afno_mlp(float* __restrict__ Fr, float* __restrict__ Fi,
              const float* __restrict__ w1r, const float* __restrict__ w1i,
              const float* __restrict__ b1,
              const float* __restrict__ w2r, const float* __restrict__ w2i,
              const float* __restrict__ b2) {
    __shared__ __align__(16) float zr[16 * LSTR];
    __shared__ __align__(16) float zi[16 * LSTR];
    __shared__ __align__(16) float hr[16 * LSTR];
    __shared__ __align__(16) float hi[16 * LSTR];

    const int blk  = blockIdx.x & (NBLK - 1);
    const int mt   = blockIdx.x >> 3;
    const int m0   = mt * 16;
    const int col0 = blk * BS;
    const int t    = threadIdx.x;

    // ---- async-stage Z tile [16 x 96] (re & im planes) into LDS ----------
    for (int idx = t; idx < 16 * (BS / 4); idx += 192) {
        const int r  = idx / (BS / 4);
        const int c4 = (idx % (BS / 4)) << 2;
        const size_t g = (size_t)(m0 + r) * Dd + col0 + c4;
        async_ld_b128(lds_off(&zr[r * LSTR + c4]), Fr + g);
        async_ld_b128(lds_off(&zi[r * LSTR + c4]), Fi + g);
    }
    async_wait0();
    __syncthreads();

    const int wave = t >> 5;           // 0..5 -> N tile
    const int lane = t & 31;
    const int hsel = lane >> 4;        // which K-half / M-half per layouts
    const int l15  = lane & 15;
    const int o0   = wave * 16;
    const int orow = o0 + l15;         // W row / bias row this lane serves

    // hint the layer-2 weights toward L2 while we chew on layer 1
    __builtin_prefetch(&w2r[orow * BS], 0, 1);
    __builtin_prefetch(&w2i[orow * BS], 0, 1);

    // ---- layer 1: h = CReLU(Z W1^T + b1) ---------------------------------
    v8f cr, ci;
    {
        const float brv = b1[orow * 2 + 0];
        const float biv = b1[orow * 2 + 1];
#pragma unroll
        for (int v = 0; v < 8; ++v) { cr[v] = brv; ci[v] = biv; }
    }
#pragma unroll 4
    for (int k0 = 0; k0 < BS; k0 += 4) {
        const int ka = k0 + 2 * hsel;
        const v2f aR = *(const v2f*)&zr[l15 * LSTR + ka];
        const v2f aI = *(const v2f*)&zi[l15 * LSTR + ka];
        const v2f bR = *(const v2f*)&w1r[orow * BS + ka];
        const v2f bI = *(const v2f*)&w1i[orow * BS + ka];
        v2f aIn; aIn.x = -aI.x; aIn.y = -aI.y;
        cr = wmma4(aR,  bR, cr);   // + Zr*W1r^T
        cr = wmma4(aIn, bI, cr);   // - Zi*W1i^T
        ci = wmma4(aR,  bI, ci);   // + Zr*W1i^T
        ci = wmma4(aI,  bR, ci);   // + Zi*W1r^T
    }
#pragma unroll
    for (int v = 0; v < 8; ++v) {
        cr[v] = fmaxf(cr[v], 0.0f);
        ci[v] = fmaxf(ci[v], 0.0f);
    }

    // ---- re-stage h through LDS (C layout -> A layout) -------------------
#pragma unroll
    for (int v = 0; v < 8; ++v) {
        const int M = v + 8 * hsel;
        hr[M * LSTR + o0 + l15] = cr[v];
        hi[M * LSTR + o0 + l15] = ci[v];
    }
    __syncthreads();

    // ---- layer 2: y = h W2^T + b2, then softshrink -----------------------
    v8f yr, yi;
    {
        const float brv = b2[orow * 2 + 0];
        const float biv = b2[orow * 2 + 1];
#pragma unroll
        for (int v = 0; v < 8; ++v) { yr[v] = brv; yi[v] = biv; }
    }
#pragma unroll 4
    for (int k0 = 0; k0 < BS; k0 += 4) {
        const int ka = k0 + 2 * hsel;
        const v2f aR = *(const v2f*)&hr[l15 * LSTR + ka];
        const v2f aI = *(const v2f*)&hi[l15 * LSTR + ka];
        const v2f bR = *(const v2f*)&w2r[orow * BS + ka];
        const v2f bI = *(const v2f*)&w2i[orow * BS + ka];
        v2f aIn; aIn.x = -aI.x; aIn.y = -aI.y;
        yr = wmma4(aR,  bR, yr);
        yr = wmma4(aIn, bI, yr);
        yi = wmma4(aR,  bI, yi);
        yi = wmma4(aI,  bR, yi);
    }

    // componentwise softshrink + in-place store
#pragma unroll
    for (int v = 0; v < 8; ++v) {
        float a = yr[v];
        a = (a > THRESH) ? (a - THRESH) : ((a < -THRESH) ? (a + THRESH) : 0.0f);
        float b = yi[v];
        b = (b > THRESH) ? (b - THRESH) : ((b < -THRESH) ? (b + THRESH) : 0.0f);
        const int M = v + 8 * hsel;
        Fr[(size_t)(m0 + M) * Dd + col0 + o0 + l15] = a;
        Fi[(size_t)(m0 + M) * Dd + col0 + o0 + l15] = b;
    }
}

// ---------------------------------------------------------------------------
// Stage E: inverse rFFT along W via Hermitian reconstruction, ortho scale,
// add identity skip, write output.  grid = (B*H) * NDG, 256 threads.
// (Load kept on the VALU path: the mirror needs the -im negate.)
// ---------------------------------------------------------------------------
__global__ __launch_bounds__(256)
void ifft_w_skip(const float* __restrict__ Fr, const float* __restrict__ Fi,
                 const float* __restrict__ x, float* __restrict__ out) {
    __shared__ __align__(16) float sr[Ww * CG];
    __shared__ __align__(16) float si[Ww * CG];
    const int wg = blockIdx.x;
    const int dg = wg % NDG;
    const int bh = wg / NDG;
    const int t  = threadIdx.x;
    const size_t fbase = (size_t)bh * Fw * Dd + dg * CG;
    for (int idx = t; idx < Fw * CG; idx += 256) {
        const int f = idx >> 5, c = idx & (CG - 1);
        const float vr = Fr[fbase + (size_t)f * Dd + c];
        const float vi = Fi[fbase + (size_t)f * Dd + c];
        const int fb = __brev((unsigned)f) >> 25;
        sr[fb * CG + c] = vr;
        si[fb * CG + c] = vi;
        if (f >= 1 && f <= 63) {                 // Hermitian mirror
            const int w2 = Ww - f;
            const int wb = __brev((unsigned)w2) >> 25;
            sr[wb * CG + c] = vr;
            si[wb * CG + c] = -vi;
        }
    }
    fft_core(sr, si, t, 1.0f);
    const float* xs = x   + (size_t)bh * Ww * Dd + dg * CG;
    float*       os = out + (size_t)bh * Ww * Dd + dg * CG;
    for (int idx = t; idx < Ww * CG; idx += 256) {
        const int w = idx >> 5, c = idx & (CG - 1);
        os[(size_t)w * Dd + c] = sr[w * CG + c] * INV128 + xs[(size_t)w * Dd + c];
    }
}

// ---------------------------------------------------------------------------
extern "C" void kernel_launch(void* const* d_in, const int* in_sizes, int n_in,
                              void* d_out, int out_size, void* d_ws, size_t ws_size,
                              hipStream_t stream) {
    const float* x   = (const float*)d_in[0];
    const float* w1r = (const float*)d_in[1];
    const float* w1i = (const float*)d_in[2];
    const float* b1  = (const float*)d_in[3];
    const float* w2r = (const float*)d_in[4];
    const float* w2i = (const float*)d_in[5];
    const float* b2  = (const float*)d_in[6];
    float* out = (float*)d_out;

    float* Fr = (float*)d_ws;                          // [TOTAL_M, D] re plane
    float* Fi = Fr + (size_t)TOTAL_M * Dd;             // [TOTAL_M, D] im plane

    fft_w_fwd  <<<Bz * Hh * NDG, 256, 0, stream>>>(x, Fr, Fi);
    fft_h      <<<Bz * Fw * NDG, 256, 0, stream>>>(Fr, Fi, -1.0f);
    afno_mlp   <<<(TOTAL_M / 16) * NBLK, 192, 0, stream>>>(Fr, Fi, w1r, w1i, b1,
                                                           w2r, w2i, b2);
    fft_h      <<<Bz * Fw * NDG, 256, 0, stream>>>(Fr, Fi, 1.0f);
    ifft_w_skip<<<Bz * Hh * NDG, 256, 0, stream>>>(Fr, Fi, x, out);
}